// GAT_50577534878113
// MI455X (gfx1250) — compile-verified
//
#include <hip/hip_runtime.h>
#include <hip/hip_bf16.h>
#include <math.h>

#define NN 50000
#define NE 800000
#define IN_CH 128
#define HIDDEN 32
#define HEADS 4
#define OUT_CH 40
#define HC 128            // HEADS*HIDDEN
#define NEG_SLOPE 0.2f
#define EPS_DEN 1e-16f
#define ENC_NEGINF 0x007FFFFFu   // enc(-inf) for orderable-uint float max

typedef __attribute__((ext_vector_type(2))) float v2f;
typedef __attribute__((ext_vector_type(8))) float v8f;

// ---------- utility: order-preserving float<->uint for atomicMax ----------
__device__ __forceinline__ unsigned fenc(float f) {
    unsigned u = __float_as_uint(f);
    return (u & 0x80000000u) ? ~u : (u | 0x80000000u);
}
__device__ __forceinline__ float fdec(unsigned u) {
    u = (u & 0x80000000u) ? (u & 0x7fffffffu) : ~u;
    return __uint_as_float(u);
}
__device__ __forceinline__ float lrelu(float v) { return v >= 0.f ? v : NEG_SLOPE * v; }

// ---------- fills ----------
__global__ void fill_f32(float* __restrict__ p, float v, int n) {
    int t = blockIdx.x * blockDim.x + threadIdx.x;
    if (t < n) p[t] = v;
}
__global__ void fill_u32(unsigned* __restrict__ p, unsigned v, int n) {
    int t = blockIdx.x * blockDim.x + threadIdx.x;
    if (t < n) p[t] = v;
}

// ---------- fp32 WMMA GEMM: C[M,Ncols] = A[M,K] @ B[K,Ncols] ----------
// wave-per-tile; M % 16 == 0, K % 4 == 0; Ncols may be ragged.
// Ragged columns are handled branch-free (clamped index + 0/1 mask) so EXEC
// stays all-1s through the WMMA loop and no per-iteration exec save/restore
// is generated.
__global__ void wmma_gemm_f32(const float* __restrict__ A, const float* __restrict__ B,
                              float* __restrict__ C, int Mtiles, int K, int Ncols, int Ntiles) {
    int lane = threadIdx.x & 31;
    int wave = threadIdx.x >> 5;
    int tid  = blockIdx.x * (blockDim.x >> 5) + wave;   // wave-uniform
    if (tid >= Mtiles * Ntiles) return;                 // uniform per wave -> EXEC stays full
    int tm = tid / Ntiles;
    int tn = tid % Ntiles;

    int row = lane & 15;   // M row for A-frag, N col for B/C-frag
    int hi  = lane >> 4;   // selects K pair {0,1} vs {2,3}

    int   bn   = tn * 16 + row;
    bool  bok  = (bn < Ncols);
    int   bnc  = bok ? bn : (Ncols - 1);   // clamped, always-valid column
    float mask = bok ? 1.f : 0.f;

    const float* ap = A + ((size_t)(tm * 16 + row)) * K;
    const float* bp = B + bnc;

    v8f acc = {};
    #pragma unroll 4
    for (int kk = 0; kk < K; kk += 4) {
        int k0 = kk + hi * 2;
        v2f a, b;
        a.x = ap[k0];
        a.y = ap[k0 + 1];
        b.x = bp[(size_t)k0 * Ncols] * mask;        // unconditional coalesced load
        b.y = bp[(size_t)(k0 + 1) * Ncols] * mask;
        acc = __builtin_amdgcn_wmma_f32_16x16x4_f32(
            /*neg_a=*/false, a, /*neg_b=*/false, b,
            /*c_mod=*/(short)0, acc, /*reuse_a=*/false, /*reuse_b=*/false);
    }

    if (bok) {
        // D layout: VGPR r -> M = r (lanes 0-15), M = r+8 (lanes 16-31); N = lane&15
        #pragma unroll
        for (int r = 0; r < 8; ++r) {
            int m = tm * 16 + r + hi * 8;
            C[(size_t)m * Ncols + bn] = acc[r];
        }
    }
}

// ---------- attention dot products: a_s[n,h] = sum_c h[n,h,c]*att_s[h,c] ----------
__global__ void att_dot_kernel(const float* __restrict__ h, const float* __restrict__ att_s,
                               const float* __restrict__ att_d, float* __restrict__ a_s,
                               float* __restrict__ a_d, int total, int H, int C) {
    int t = blockIdx.x * blockDim.x + threadIdx.x;
    if (t >= total) return;
    int hh = t % H;
    int n  = t / H;
    const float* hp = h + (size_t)n * H * C + (size_t)hh * C;
    const float* sp = att_s + hh * C;
    const float* dp = att_d + hh * C;
    float ss = 0.f, sd = 0.f;
    for (int c = 0; c < C; ++c) { float v = hp[c]; ss += v * sp[c]; sd += v * dp[c]; }
    a_s[t] = ss;
    a_d[t] = sd;
}

// ---------- edge pass 1: segment max of leaky-relu logits ----------
__global__ void edge_max_kernel(const int* __restrict__ ei, const float* __restrict__ a_s,
                                const float* __restrict__ a_d, unsigned* __restrict__ segmax,
                                int E, int H) {
    int t = blockIdx.x * blockDim.x + threadIdx.x;
    if (t >= E * H) return;
    int e = t / H, hh = t % H;
    int s = ei[e], d = ei[E + e];
    float v = lrelu(a_s[s * H + hh] + a_d[d * H + hh]);
    atomicMax(&segmax[d * H + hh], fenc(v));
}

// ---------- edge pass 2: denom += exp(e - max) ----------
__global__ void edge_denom_kernel(const int* __restrict__ ei, const float* __restrict__ a_s,
                                  const float* __restrict__ a_d, const unsigned* __restrict__ segmax,
                                  float* __restrict__ denom, int E, int H) {
    int t = blockIdx.x * blockDim.x + threadIdx.x;
    if (t >= E * H) return;
    int e = t / H, hh = t % H;
    int s = ei[e], d = ei[E + e];
    int idx = d * H + hh;
    float v = lrelu(a_s[s * H + hh] + a_d[idx]);
    atomicAdd(&denom[idx], expf(v - fdec(segmax[idx])));
}

// ---------- edge pass 3 (layer 1): wave per edge, 128 channels, H=4 C=32 ----------
__global__ void scatter1_kernel(const int* __restrict__ ei, const float* __restrict__ a_s,
                                const float* __restrict__ a_d, const unsigned* __restrict__ segmax,
                                const float* __restrict__ denom, const float* __restrict__ h1,
                                float* __restrict__ acc, int E) {
    int wid  = (blockIdx.x * blockDim.x + threadIdx.x) >> 5;
    int lane = threadIdx.x & 31;
    if (wid >= E) return;
    int s = ei[wid], d = ei[E + wid];
    const float* hs = h1 + (size_t)s * HC;
    float* ap = acc + (size_t)d * HC;
    #pragma unroll
    for (int j = 0; j < HEADS; ++j) {
        int idx = d * HEADS + j;
        float v = lrelu(a_s[s * HEADS + j] + a_d[idx]);
        float alpha = expf(v - fdec(segmax[idx])) / (denom[idx] + EPS_DEN);
        int c = j * HIDDEN + lane;                 // HIDDEN==32 -> lane covers one head
        atomicAdd(&ap[c], hs[c] * alpha);
    }
}

// ---------- ELU in place ----------
__global__ void elu_kernel(float* __restrict__ p, int n) {
    int t = blockIdx.x * blockDim.x + threadIdx.x;
    if (t >= n) return;
    float x = p[t];
    p[t] = x > 0.f ? x : expm1f(x);
}

// ---------- edge pass 3 (layer 2): wave per edge, 40 channels, H=1 ----------
__global__ void scatter2_kernel(const int* __restrict__ ei, const float* __restrict__ a_s,
                                const float* __restrict__ a_d, const unsigned* __restrict__ segmax,
                                const float* __restrict__ denom, const float* __restrict__ g2,
                                float* __restrict__ out, int E) {
    int wid  = (blockIdx.x * blockDim.x + threadIdx.x) >> 5;
    int lane = threadIdx.x & 31;
    if (wid >= E) return;
    int s = ei[wid], d = ei[E + wid];
    float v = lrelu(a_s[s] + a_d[d]);
    float alpha = expf(v - fdec(segmax[d])) / (denom[d] + EPS_DEN);
    const float* gs = g2 + (size_t)s * OUT_CH;
    float* op = out + (size_t)d * OUT_CH;
    if (lane < OUT_CH) atomicAdd(&op[lane], gs[lane] * alpha);
    int c2 = lane + 32;
    if (c2 < OUT_CH) atomicAdd(&op[c2], gs[c2] * alpha);
}

static inline int cdiv(long long a, long long b) { return (int)((a + b - 1) / b); }

extern "C" void kernel_launch(void* const* d_in, const int* in_sizes, int n_in,
                              void* d_out, int out_size, void* d_ws, size_t ws_size,
                              hipStream_t stream) {
    const float* x        = (const float*)d_in[0];
    const int*   ei       = (const int*)d_in[1];    // [2, NE]
    const float* W1       = (const float*)d_in[2];  // [128, 128]
    const float* att_s1   = (const float*)d_in[3];  // [4, 32]
    const float* att_d1   = (const float*)d_in[4];
    const float* W2       = (const float*)d_in[5];  // [128, 40]
    const float* att_s2   = (const float*)d_in[6];  // [1, 40]
    const float* att_d2   = (const float*)d_in[7];
    float* out = (float*)d_out;                     // [NN, 40]

    // workspace layout (floats)
    float* ws = (float*)d_ws;
    size_t off = 0;
    float*    h1   = ws + off; off += (size_t)NN * HC;     // x @ W1
    float*    as1  = ws + off; off += (size_t)NN * HEADS;
    float*    ad1  = ws + off; off += (size_t)NN * HEADS;
    unsigned* sm1  = (unsigned*)(ws + off); off += (size_t)NN * HEADS;
    float*    dn1  = ws + off; off += (size_t)NN * HEADS;
    float*    acc1 = ws + off; off += (size_t)NN * HC;     // layer-1 output, then ELU'd
    float*    g2   = ws + off; off += (size_t)NN * OUT_CH; // acc1 @ W2
    float*    as2  = ws + off; off += (size_t)NN;
    float*    ad2  = ws + off; off += (size_t)NN;
    unsigned* sm2  = (unsigned*)(ws + off); off += (size_t)NN;
    float*    dn2  = ws + off; off += (size_t)NN;

    const int B = 256;

    // ---- init accumulators / maxima ----
    fill_f32<<<cdiv((long long)NN * HC, B), B, 0, stream>>>(acc1, 0.f, NN * HC);
    fill_f32<<<cdiv(NN * HEADS, B), B, 0, stream>>>(dn1, 0.f, NN * HEADS);
    fill_u32<<<cdiv(NN * HEADS, B), B, 0, stream>>>(sm1, ENC_NEGINF, NN * HEADS);
    fill_f32<<<cdiv(NN, B), B, 0, stream>>>(dn2, 0.f, NN);
    fill_u32<<<cdiv(NN, B), B, 0, stream>>>(sm2, ENC_NEGINF, NN);
    fill_f32<<<cdiv((long long)NN * OUT_CH, B), B, 0, stream>>>(out, 0.f, NN * OUT_CH);

    // ---- layer 1 ----
    {   // GEMM1: [50000,128] @ [128,128]; 3125 M-tiles x 8 N-tiles, 8 waves/block
        int Mtiles = NN / 16, Ntiles = HC / 16;
        int waves = Mtiles * Ntiles;
        wmma_gemm_f32<<<cdiv(waves, 8), B, 0, stream>>>(x, W1, h1, Mtiles, IN_CH, HC, Ntiles);
    }
    att_dot_kernel<<<cdiv(NN * HEADS, B), B, 0, stream>>>(h1, att_s1, att_d1, as1, ad1,
                                                          NN * HEADS, HEADS, HIDDEN);
    edge_max_kernel<<<cdiv((long long)NE * HEADS, B), B, 0, stream>>>(ei, as1, ad1, sm1, NE, HEADS);
    edge_denom_kernel<<<cdiv((long long)NE * HEADS, B), B, 0, stream>>>(ei, as1, ad1, sm1, dn1, NE, HEADS);
    scatter1_kernel<<<cdiv((long long)NE * 32, B), B, 0, stream>>>(ei, as1, ad1, sm1, dn1, h1, acc1, NE);
    elu_kernel<<<cdiv((long long)NN * HC, B), B, 0, stream>>>(acc1, NN * HC);

    // ---- layer 2 ----
    {   // GEMM2: [50000,128] @ [128,40]; 3 ragged N-tiles
        int Mtiles = NN / 16, Ntiles = (OUT_CH + 15) / 16;
        int waves = Mtiles * Ntiles;
        wmma_gemm_f32<<<cdiv(waves, 8), B, 0, stream>>>(acc1, W2, g2, Mtiles, HC, OUT_CH, Ntiles);
    }
    att_dot_kernel<<<cdiv(NN, B), B, 0, stream>>>(g2, att_s2, att_d2, as2, ad2, NN, 1, OUT_CH);
    edge_max_kernel<<<cdiv(NE, B), B, 0, stream>>>(ei, as2, ad2, sm2, NE, 1);
    edge_denom_kernel<<<cdiv(NE, B), B, 0, stream>>>(ei, as2, ad2, sm2, dn2, NE, 1);
    scatter2_kernel<<<cdiv((long long)NE * 32, B), B, 0, stream>>>(ei, as2, ad2, sm2, dn2, g2, out, NE);
}